// RNNAttentionNaive_5368709120364
// MI455X (gfx1250) — compile-verified
//
#include <hip/hip_runtime.h>
#include <hip/hip_bf16.h>

// Problem constants (match reference)
#define BB     8
#define T_OUT  256
#define T_IN   512
#define LL     128
#define G4     512   // 4*L

typedef __attribute__((ext_vector_type(2))) float v2f;
typedef __attribute__((ext_vector_type(8))) float v8f;

// D = A(16x4) x B(4x16) + C, fp32 WMMA (CDNA5)
__device__ __forceinline__ v8f wmma_k4(v2f a, v2f b, v8f c) {
    return __builtin_amdgcn_wmma_f32_16x16x4_f32(
        /*neg_a=*/false, a, /*neg_b=*/false, b,
        /*c_mod=*/(short)0, c, /*reuse_a=*/false, /*reuse_b=*/false);
}

__device__ __forceinline__ float sigf(float x) { return 1.0f / (1.0f + __expf(-x)); }

// ---------------------------------------------------------------------------
// Generic fp32 WMMA GEMM:  C[M,N] = A[M,K] @ W[K,N] + bias[N]
// One wave per 16x16 output tile; 256-thread blocks = 8 tiles/block.
// M, N multiples of 16; K multiple of 4; tile count multiple of 8.
// ---------------------------------------------------------------------------
__global__ void __launch_bounds__(256)
gemm_wmma_f32(const float* __restrict__ A, int lda,
              const float* __restrict__ W, int ldw,
              const float* __restrict__ bias,
              float* __restrict__ C, int ldc,
              int M, int N, int K)
{
    const int wave = threadIdx.x >> 5;
    const int lane = threadIdx.x & 31;
    const int m16  = lane & 15;
    const int half = lane >> 4;

    const int tiles_n = N >> 4;
    const int tile    = blockIdx.x * 8 + wave;
    if (tile >= (M >> 4) * tiles_n) return;
    const int tile_m = tile / tiles_n;
    const int tile_n = tile - tile_m * tiles_n;

    const int n = tile_n * 16 + m16;

    v8f acc = {};
    const float bv = bias ? bias[n] : 0.0f;
#pragma unroll
    for (int v = 0; v < 8; ++v) acc[v] = bv;

    const float* __restrict__ Arow = A + (size_t)(tile_m * 16 + m16) * lda;
#pragma unroll 4
    for (int k0 = 0; k0 < K; k0 += 4) {
        const int ka = k0 + 2 * half;
        v2f af, bf;
        af[0] = Arow[ka];
        af[1] = Arow[ka + 1];
        bf[0] = W[(size_t)ka * ldw + n];
        bf[1] = W[(size_t)(ka + 1) * ldw + n];
        acc = wmma_k4(af, bf, acc);
    }

#pragma unroll
    for (int v = 0; v < 8; ++v) {
        const int row = tile_m * 16 + v + 8 * half;
        C[(size_t)row * ldc + n] = acc[v];
    }
}

// ---------------------------------------------------------------------------
// LSTM scan: single workgroup (512 threads = 16 wave32), all 8 batches.
// Per step: z(8x512) = xz[:,t,:] + h(8x128) @ Wr(128x512) via WMMA with the
// 8-row h matrix zero-padded to 16 rows; gates in VALU/TRANS.
// Writes hidden sequence into d_out[:, :, 0:128] and final h,c to the tail.
// ---------------------------------------------------------------------------
__global__ void __launch_bounds__(512)
lstm_scan(const float* __restrict__ xz,   // (B*T_OUT, 512)
          const float* __restrict__ Wr,   // (128, 512)
          float* __restrict__ out)        // full d_out
{
    __shared__ float hbuf[16][LL];   // rows 8..15 stay zero (WMMA padding)
    __shared__ float cbuf[BB][LL];
    __shared__ float zbuf[16][G4];

    const int tid = threadIdx.x;
    for (int i = tid; i < 16 * LL; i += 512) hbuf[i >> 7][i & 127] = 0.0f;
    for (int i = tid; i < BB * LL; i += 512) cbuf[i >> 7][i & 127] = 0.0f;
    __syncthreads();

    const int wave = tid >> 5;
    const int lane = tid & 31;
    const int m16  = lane & 15;
    const int half = lane >> 4;

    for (int t = 0; t < T_OUT; ++t) {
        // ---- z = xz_t + h @ Wr : 32 N-tiles, 2 per wave ----
#pragma unroll
        for (int tt = 0; tt < 2; ++tt) {
            const int n = (wave * 2 + tt) * 16 + m16;
            v8f acc = {};
#pragma unroll
            for (int v = 0; v < 8; ++v) {
                const int row = v + 8 * half;               // batch index
                acc[v] = (row < BB) ? xz[((size_t)row * T_OUT + t) * G4 + n] : 0.0f;
            }
#pragma unroll 4
            for (int k0 = 0; k0 < LL; k0 += 4) {
                const int ka = k0 + 2 * half;
                v2f af, bf;
                af[0] = hbuf[m16][ka];
                af[1] = hbuf[m16][ka + 1];
                bf[0] = Wr[(size_t)ka * G4 + n];
                bf[1] = Wr[(size_t)(ka + 1) * G4 + n];
                acc = wmma_k4(af, bf, acc);
            }
#pragma unroll
            for (int v = 0; v < 8; ++v) zbuf[v + 8 * half][n] = acc[v];
        }
        __syncthreads();

        // ---- gates: 8 batches x 128 cells = 1024 updates, 2 per thread ----
#pragma unroll
        for (int r = 0; r < 2; ++r) {
            const int idx  = tid + 512 * r;
            const int b    = idx >> 7;
            const int cell = idx & 127;
            const float zi = zbuf[b][cell];
            const float zf = zbuf[b][LL + cell];
            const float zg = zbuf[b][2 * LL + cell];
            const float zo = zbuf[b][3 * LL + cell];
            const float cn = sigf(zf) * cbuf[b][cell] + sigf(zi) * tanhf(zg);
            const float hn = sigf(zo) * tanhf(cn);
            cbuf[b][cell] = cn;
            hbuf[b][cell] = hn;
            out[((size_t)b * T_OUT + t) * 256 + cell] = hn;   // x part of concat
        }
        __syncthreads();
    }

    // final states: tuple tail (h then c)
    const size_t tail = (size_t)BB * T_OUT * 256;
#pragma unroll
    for (int r = 0; r < 2; ++r) {
        const int idx  = tid + 512 * r;
        const int b    = idx >> 7;
        const int cell = idx & 127;
        out[tail + (size_t)b * LL + cell]           = hbuf[b][cell];
        out[tail + BB * LL + (size_t)b * LL + cell] = cbuf[b][cell];
    }
}

// ---------------------------------------------------------------------------
// Bahdanau attention: one block per (batch, 16-query tile).
// Phase A: scores via LDS-staged key tiles + tanh dot (TRANS-bound).
// Phase B: wave32 softmax over T_IN=512 via __shfl_xor.
// Phase C: weighted = P(16x512) @ attended(512x128) via WMMA, written into
//          the concat slot d_out[:, :, 128:256].
// ---------------------------------------------------------------------------
__global__ void __launch_bounds__(256)
attention(const float* __restrict__ keys,      // (B*T_IN, 128)
          const float* __restrict__ q,         // (B*T_OUT, 128)
          const float* __restrict__ attended,  // (B*T_IN, 128)
          const float* __restrict__ W3,        // (128)
          const float* __restrict__ b3,        // (1)
          float* __restrict__ out)
{
    __shared__ float S[16][T_IN];     // scores -> probabilities
    __shared__ float tq[16][LL];
    __shared__ float kbuf[16][LL];
    __shared__ float w3s[LL];

    const int b  = blockIdx.x >> 4;
    const int q0 = (blockIdx.x & 15) * 16;
    const int tid = threadIdx.x;

    for (int i = tid; i < 16 * LL; i += 256)
        tq[i >> 7][i & 127] = q[((size_t)b * T_OUT + q0 + (i >> 7)) * LL + (i & 127)];
    if (tid < LL) w3s[tid] = W3[tid];
    __syncthreads();
    const float b3v = b3[0];

    // ---- Phase A: scores ----
    const int sm = tid >> 4;   // query row within tile
    const int sk = tid & 15;   // key row within staged tile
    for (int kt = 0; kt < T_IN / 16; ++kt) {
        for (int i = tid; i < 16 * LL; i += 256)
            kbuf[i >> 7][i & 127] = keys[((size_t)b * T_IN + kt * 16 + (i >> 7)) * LL + (i & 127)];
        __syncthreads();
        float s = 0.0f;
#pragma unroll 4
        for (int l = 0; l < LL; ++l)
            s += tanhf(kbuf[sk][l] + tq[sm][l]) * w3s[l];
        S[sm][kt * 16 + sk] = s + b3v;
        __syncthreads();
    }

    // ---- Phase B: softmax per query row (8 waves x 2 rows) ----
    const int wave = tid >> 5;
    const int lane = tid & 31;
#pragma unroll
    for (int r = 0; r < 2; ++r) {
        const int m = wave * 2 + r;
        float mx = -3.402823466e38f;
        for (int k = lane; k < T_IN; k += 32) mx = fmaxf(mx, S[m][k]);
#pragma unroll
        for (int off = 16; off > 0; off >>= 1) mx = fmaxf(mx, __shfl_xor(mx, off, 32));
        float sum = 0.0f;
        for (int k = lane; k < T_IN; k += 32) {
            const float e = __expf(S[m][k] - mx);
            S[m][k] = e;
            sum += e;
        }
#pragma unroll
        for (int off = 16; off > 0; off >>= 1) sum += __shfl_xor(sum, off, 32);
        const float inv = 1.0f / sum;
        for (int k = lane; k < T_IN; k += 32) S[m][k] *= inv;
    }
    __syncthreads();

    // ---- Phase C: weighted = P @ attended, one 16-wide N-tile per wave ----
    const int m16  = lane & 15;
    const int half = lane >> 4;
    const int n    = wave * 16 + m16;
    v8f acc = {};
#pragma unroll 4
    for (int k0 = 0; k0 < T_IN; k0 += 4) {
        const int ka = k0 + 2 * half;
        v2f af, bf;
        af[0] = S[m16][ka];
        af[1] = S[m16][ka + 1];
        bf[0] = attended[((size_t)b * T_IN + ka) * LL + n];
        bf[1] = attended[((size_t)b * T_IN + ka + 1) * LL + n];
        acc = wmma_k4(af, bf, acc);
    }
#pragma unroll
    for (int v = 0; v < 8; ++v) {
        const int m = v + 8 * half;
        out[((size_t)b * T_OUT + q0 + m) * 256 + LL + n] = acc[v];
    }
}

// ---------------------------------------------------------------------------
extern "C" void kernel_launch(void* const* d_in, const int* in_sizes, int n_in,
                              void* d_out, int out_size, void* d_ws, size_t ws_size,
                              hipStream_t stream)
{
    const float* inputs   = (const float*)d_in[0];   // (B, T_OUT, 128)
    const float* attended = (const float*)d_in[1];   // (B, T_IN, 128)
    const float* Wk       = (const float*)d_in[2];   // (128, 512)
    const float* Wr       = (const float*)d_in[3];   // (128, 512)
    const float* bl       = (const float*)d_in[4];   // (512)
    const float* W1       = (const float*)d_in[5];   // (128, 128)
    const float* b1       = (const float*)d_in[6];   // (128)
    const float* W2       = (const float*)d_in[7];   // (128, 128)
    const float* b2       = (const float*)d_in[8];   // (128)
    const float* W3       = (const float*)d_in[9];   // (128, 1)
    const float* b3       = (const float*)d_in[10];  // (1)

    float* out = (float*)d_out;
    float* ws  = (float*)d_ws;
    float* keys = ws;                                   // 4096*128
    float* xz   = keys + (size_t)BB * T_IN * LL;        // 2048*512
    float* q    = xz + (size_t)BB * T_OUT * G4;         // 2048*128

    // keys = attended @ W1 + b1   (M=4096, N=128, K=128) -> 2048 tiles / 8
    gemm_wmma_f32<<<256, 256, 0, stream>>>(attended, LL, W1, LL, b1,
                                           keys, LL, BB * T_IN, LL, LL);

    // xz = inputs @ Wk + bias     (M=2048, N=512, K=128) -> 4096 tiles / 8
    gemm_wmma_f32<<<512, 256, 0, stream>>>(inputs, LL, Wk, G4, bl,
                                           xz, G4, BB * T_OUT, G4, LL);

    // sequential LSTM scan (writes x into d_out[:,:,0:128] and h,c tail)
    lstm_scan<<<1, 512, 0, stream>>>(xz, Wr, out);

    // q = x @ W2 + b2 (x read from d_out with row stride 256)
    gemm_wmma_f32<<<128, 256, 0, stream>>>(out, 256, W2, LL, b2,
                                           q, LL, BB * T_OUT, LL, LL);

    // attention + weighted sum into d_out[:,:,128:256]
    attention<<<BB * (T_OUT / 16), 256, 0, stream>>>(keys, q, attended, W3, b3, out);
}